// CustomAttentionLayer_40303973106206
// MI455X (gfx1250) — compile-verified
//
#include <hip/hip_runtime.h>
#include <math.h>

#define K_CL 1024
#define DMODEL 256
#define NPIX 65536

typedef float v2f __attribute__((ext_vector_type(2)));
typedef float v4f __attribute__((ext_vector_type(4)));
typedef float v8f __attribute__((ext_vector_type(8)));

// ---------------------------------------------------------------- zero
__global__ __launch_bounds__(256) void zero_f32(float* __restrict__ p, int n) {
  int i = blockIdx.x * 256 + threadIdx.x;
  if (i < n) p[i] = 0.f;
}

// ---------------------------------------------------------------- kernel 1:
// acc[k][0..15] += { a*x, a*y, a*(iso==1), ..., a*(iso==12), 0, 0 }
// lanes stride over K (float4/thread); iso class is row-uniform -> SGPR switch.
// A is streamed once (512 MB total) -> non-temporal loads to spare L2.
__global__ __launch_bounds__(256) void reduce_assign(
    const float* __restrict__ A, const float* __restrict__ coords,
    const int* __restrict__ iso, float* __restrict__ acc) {
  __shared__ float sx[256], sy[256];
  __shared__ int sc[256];
  const int t = threadIdx.x;
  const int kb = t * 4;
  float ax0 = 0, ax1 = 0, ax2 = 0, ax3 = 0;
  float ay0 = 0, ay1 = 0, ay2 = 0, ay3 = 0;
  float ai[12][4];
#pragma unroll
  for (int c = 0; c < 12; c++) { ai[c][0] = 0; ai[c][1] = 0; ai[c][2] = 0; ai[c][3] = 0; }
  const int ROWS = NPIX / gridDim.x;
  const int n0 = blockIdx.x * ROWS;
  for (int ch = 0; ch < ROWS; ch += 256) {
    __syncthreads();
    int n = n0 + ch + t;
    sx[t] = coords[2 * n];
    sy[t] = coords[2 * n + 1];
    sc[t] = iso[n] - 1;
    __syncthreads();
#pragma unroll 2
    for (int r = 0; r < 256; r++) {
      const v4f a = __builtin_nontemporal_load(
          (const v4f*)(A + (size_t)(n0 + ch + r) * K_CL + kb));
      float x = sx[r], y = sy[r];
      int c = __builtin_amdgcn_readfirstlane(sc[r]);
      ax0 += a.x * x; ax1 += a.y * x; ax2 += a.z * x; ax3 += a.w * x;
      ay0 += a.x * y; ay1 += a.y * y; ay2 += a.z * y; ay3 += a.w * y;
#define ISO_CASE(C) case C: ai[C][0]+=a.x; ai[C][1]+=a.y; ai[C][2]+=a.z; ai[C][3]+=a.w; break;
      switch (c) {
        ISO_CASE(0) ISO_CASE(1) ISO_CASE(2) ISO_CASE(3)
        ISO_CASE(4) ISO_CASE(5) ISO_CASE(6) ISO_CASE(7)
        ISO_CASE(8) ISO_CASE(9) ISO_CASE(10) ISO_CASE(11)
        default: break;
      }
#undef ISO_CASE
    }
  }
  float axs[4] = {ax0, ax1, ax2, ax3};
  float ays[4] = {ay0, ay1, ay2, ay3};
#pragma unroll
  for (int q = 0; q < 4; q++) {
    float* p = acc + (size_t)(kb + q) * 16;
    atomicAdd(p + 0, axs[q]);
    atomicAdd(p + 1, ays[q]);
#pragma unroll
    for (int c = 0; c < 12; c++) atomicAdd(p + 2 + c, ai[c][q]);
  }
}

// ---------------------------------------------------------------- kernel 2:
// centroids, sensor distances, and the separated MLP halves:
//   uT[h][i] = d_k[i]@g1_w[0:12,h] + g1_b[h],  vT[h][j] = d_kp[j]@g1_w[12:24,h]
//   hbiT[h][i] = n_ks[i]*h1_w[1,h] + h1_b[h],  hbjT[h][j] = n_kps[j]*h1_w[2,h]
__global__ __launch_bounds__(256) void finalize_clusters(
    const float* __restrict__ accq, const float* __restrict__ acck,
    const float* __restrict__ sensor,
    const float* __restrict__ g1w, const float* __restrict__ g1b,
    const float* __restrict__ h1w, const float* __restrict__ h1b,
    float* __restrict__ qc, float* __restrict__ kc,
    float* __restrict__ uT, float* __restrict__ vT,
    float* __restrict__ hbiT, float* __restrict__ hbjT) {
  int k = blockIdx.x * 256 + threadIdx.x;
  float sxx = sensor[0], syy = sensor[1];
  { // q side
    const float* p = accq + (size_t)k * 16;
    float d[12]; float as = 0.f;
#pragma unroll
    for (int c = 0; c < 12; c++) { d[c] = p[2 + c]; as += d[c]; }
    float inv_sw = 1.f / (as + 1e-6f);
    float cx = p[0] * inv_sw, cy = p[1] * inv_sw;
    qc[2 * k] = cx; qc[2 * k + 1] = cy;
    float invs = 1.f / as;
#pragma unroll
    for (int c = 0; c < 12; c++) d[c] *= invs;
    float dx = cx - sxx, dy = cy - syy;
    float nk = sqrtf(dx * dx + dy * dy);
    for (int h = 0; h < 32; h++) {
      float u = g1b[h];
#pragma unroll
      for (int c = 0; c < 12; c++) u = fmaf(d[c], g1w[c * 32 + h], u);
      uT[h * K_CL + k] = u;
      hbiT[h * K_CL + k] = fmaf(nk, h1w[32 + h], h1b[h]);
    }
  }
  { // k side
    const float* p = acck + (size_t)k * 16;
    float d[12]; float as = 0.f;
#pragma unroll
    for (int c = 0; c < 12; c++) { d[c] = p[2 + c]; as += d[c]; }
    float inv_sw = 1.f / (as + 1e-6f);
    float cx = p[0] * inv_sw, cy = p[1] * inv_sw;
    kc[2 * k] = cx; kc[2 * k + 1] = cy;
    float invs = 1.f / as;
#pragma unroll
    for (int c = 0; c < 12; c++) d[c] *= invs;
    float dx = cx - sxx, dy = cy - syy;
    float nk = sqrtf(dx * dx + dy * dy);
    for (int h = 0; h < 32; h++) {
      float v = 0.f;
#pragma unroll
      for (int c = 0; c < 12; c++) v = fmaf(d[c], g1w[(12 + c) * 32 + h], v);
      vT[h * K_CL + k] = v;
      hbjT[h * K_CL + k] = nk * h1w[64 + h];
    }
  }
}

// ---------------------------------------------------------------- kernel 3:
// R[i][j] = relu(sum_h relu(u+v)*g2w + g2b) * relu(sum_h relu(nkk*w0+hbi+hbj)*h2w + h2b)
__global__ __launch_bounds__(256) void compute_R(
    const float* __restrict__ qc, const float* __restrict__ kc,
    const float* __restrict__ uT, const float* __restrict__ vT,
    const float* __restrict__ hbiT, const float* __restrict__ hbjT,
    const float* __restrict__ g2w, const float* __restrict__ g2b,
    const float* __restrict__ h1w, const float* __restrict__ h2w,
    const float* __restrict__ h2b, float* __restrict__ Rm) {
  __shared__ float sg2[32], sh0[32], sh2[32];
  int t = threadIdx.x;
  if (t < 32) { sg2[t] = g2w[t]; sh0[t] = h1w[t]; sh2[t] = h2w[t]; }
  __syncthreads();
  int i = blockIdx.y;
  int j = blockIdx.x * 256 + t;
  float dx = qc[2 * i] - kc[2 * j];
  float dy = qc[2 * i + 1] - kc[2 * j + 1];
  float nkk = sqrtf(dx * dx + dy * dy);
  float accg = 0.f, acch = 0.f;
#pragma unroll 4
  for (int h = 0; h < 32; h++) {
    float gh = fmaxf(uT[h * K_CL + i] + vT[h * K_CL + j], 0.f);
    accg = fmaf(gh, sg2[h], accg);
    float hh = fmaxf(fmaf(nkk, sh0[h], hbiT[h * K_CL + i] + hbjT[h * K_CL + j]), 0.f);
    acch = fmaf(hh, sh2[h], acch);
  }
  float G = fmaxf(accg + g2b[0], 0.f);
  float H = fmaxf(acch + h2b[0], 0.f);
  Rm[(size_t)i * K_CL + j] = G * H;
}

// ---------------------------------------------------------------- GEMM (f32 WMMA 16x16x4)
// One wave per 16x16 output tile. EPI: 0=plain, 1=*scale*R, 2=bias+relu, 3=bias.
template <int EPI, bool TB>
__global__ __launch_bounds__(256) void gemm_wmma(
    const float* __restrict__ A, const float* __restrict__ B, float* __restrict__ C,
    int N, int Kd, int lda, int ldb, int ldc,
    const float* __restrict__ Rm, float scale, const float* __restrict__ bias) {
  const int lane = threadIdx.x & 31;
  const int wave = threadIdx.x >> 5;
  const int tile = blockIdx.x * 8 + wave;
  const int ntn = N >> 4;
  const int tm = (tile / ntn) << 4;
  const int tn = (tile % ntn) << 4;
  const int l15 = lane & 15;
  const int lh = lane >> 4;
  v8f acc = {0.f, 0.f, 0.f, 0.f, 0.f, 0.f, 0.f, 0.f};
  const float* arow = A + (size_t)(tm + l15) * lda + 2 * lh;
  const float* brow = TB ? (B + (size_t)(tn + l15) * ldb + 2 * lh) : B;
#pragma unroll 4
  for (int kk = 0; kk < Kd; kk += 4) {
    v2f a = *(const v2f*)(arow + kk);
    v2f b;
    if (TB) {
      b = *(const v2f*)(brow + kk);
    } else {
      b.x = B[(size_t)(kk + 2 * lh) * ldb + tn + l15];
      b.y = B[(size_t)(kk + 2 * lh + 1) * ldb + tn + l15];
    }
    acc = __builtin_amdgcn_wmma_f32_16x16x4_f32(false, a, false, b, (short)0, acc,
                                                false, false);
  }
#pragma unroll
  for (int v = 0; v < 8; v++) {
    int row = tm + v + 8 * lh;
    int col = tn + l15;
    float val = acc[v] * scale;
    if (EPI == 1) val *= Rm[(size_t)row * N + col];
    if (EPI == 2) val = fmaxf(val + bias[col], 0.f);
    if (EPI == 3) val = val + bias[col];
    C[(size_t)row * ldc + col] = val;
  }
}

// ---------------------------------------------------------------- softmax rows (in place)
__global__ __launch_bounds__(256) void softmax_rows(float* __restrict__ Mat) {
  __shared__ float red[256];
  int t = threadIdx.x;
  float* p = Mat + (size_t)blockIdx.x * K_CL;
  float m = -3.402823466e38f;
  for (int j = t; j < K_CL; j += 256) m = fmaxf(m, p[j]);
  red[t] = m; __syncthreads();
  for (int s = 128; s > 0; s >>= 1) { if (t < s) red[t] = fmaxf(red[t], red[t + s]); __syncthreads(); }
  m = red[0]; __syncthreads();
  float sum = 0.f;
  for (int j = t; j < K_CL; j += 256) { float e = expf(p[j] - m); p[j] = e; sum += e; }
  red[t] = sum; __syncthreads();
  for (int s = 128; s > 0; s >>= 1) { if (t < s) red[t] += red[t + s]; __syncthreads(); }
  float inv = 1.f / red[0];
  __syncthreads();
  for (int j = t; j < K_CL; j += 256) p[j] *= inv;
}

// ---------------------------------------------------------------- out = LN(a + b)
__global__ __launch_bounds__(256) void add_layernorm(
    const float* __restrict__ a, const float* __restrict__ b,
    const float* __restrict__ g, const float* __restrict__ be,
    float* __restrict__ out) {
  __shared__ float red[256];
  int t = threadIdx.x;
  size_t idx = (size_t)blockIdx.x * DMODEL + t;
  float v = a[idx] + b[idx];
  red[t] = v; __syncthreads();
  for (int s = 128; s > 0; s >>= 1) { if (t < s) red[t] += red[t + s]; __syncthreads(); }
  float m = red[0] * (1.f / DMODEL);
  __syncthreads();
  red[t] = v * v; __syncthreads();
  for (int s = 128; s > 0; s >>= 1) { if (t < s) red[t] += red[t + s]; __syncthreads(); }
  float var = red[0] * (1.f / DMODEL) - m * m;
  out[idx] = (v - m) * rsqrtf(var + 1e-6f) * g[t] + be[t];
}

// ---------------------------------------------------------------- launch
extern "C" void kernel_launch(void* const* d_in, const int* in_sizes, int n_in,
                              void* d_out, int out_size, void* d_ws, size_t ws_size,
                              hipStream_t stream) {
  (void)in_sizes; (void)n_in; (void)out_size; (void)ws_size;
  const float* sensor = (const float*)d_in[0];
  const float* query  = (const float*)d_in[1];
  const float* key_emb= (const float*)d_in[2];
  const float* qA     = (const float*)d_in[3];
  const float* kA     = (const float*)d_in[4];
  const float* qcoord = (const float*)d_in[5];
  const float* kcoord = (const float*)d_in[6];
  const int*   qiso   = (const int*)d_in[7];
  const int*   kiso   = (const int*)d_in[8];
  const float* g1w = (const float*)d_in[9];
  const float* g1b = (const float*)d_in[10];
  const float* g2w = (const float*)d_in[11];
  const float* g2b = (const float*)d_in[12];
  const float* h1w = (const float*)d_in[13];
  const float* h1b = (const float*)d_in[14];
  const float* h2w = (const float*)d_in[15];
  const float* h2b = (const float*)d_in[16];
  const float* f1w = (const float*)d_in[17];
  const float* f1b = (const float*)d_in[18];
  const float* f2w = (const float*)d_in[19];
  const float* f2b = (const float*)d_in[20];
  const float* ln1g = (const float*)d_in[21];
  const float* ln1b = (const float*)d_in[22];
  const float* ln2g = (const float*)d_in[23];
  const float* ln2b = (const float*)d_in[24];

  float* ws = (float*)d_ws;
  size_t o = 0;
  float* accq = ws + o; o += K_CL * 16;
  float* acck = ws + o; o += K_CL * 16;
  float* qc   = ws + o; o += K_CL * 2;
  float* kc   = ws + o; o += K_CL * 2;
  float* uT   = ws + o; o += 32 * K_CL;
  float* vT   = ws + o; o += 32 * K_CL;
  float* hbiT = ws + o; o += 32 * K_CL;
  float* hbjT = ws + o; o += 32 * K_CL;
  float* Rm   = ws + o; o += (size_t)K_CL * K_CL;
  float* attn = ws + o; o += (size_t)K_CL * K_CL;
  float* aout = ws + o; o += (size_t)K_CL * DMODEL;
  float* xbuf = ws + o; o += (size_t)K_CL * DMODEL;
  float* ffnh = ws + o; o += (size_t)K_CL * 4 * DMODEL;
  float* ffno = ws + o; o += (size_t)K_CL * DMODEL;

  // 1) zero accumulators, reduce both assignment matrices (512 MB -> dominant)
  zero_f32<<<(2 * K_CL * 16 + 255) / 256, 256, 0, stream>>>(accq, 2 * K_CL * 16);
  reduce_assign<<<256, 256, 0, stream>>>(qA, qcoord, qiso, accq);
  reduce_assign<<<256, 256, 0, stream>>>(kA, kcoord, kiso, acck);

  // 2) centroids / iso encodings / separated MLP halves
  finalize_clusters<<<K_CL / 256, 256, 0, stream>>>(accq, acck, sensor, g1w, g1b,
                                                    h1w, h1b, qc, kc, uT, vT, hbiT, hbjT);

  // 3) pairwise geometric factor R (K x K)
  compute_R<<<dim3(K_CL / 256, K_CL), 256, 0, stream>>>(qc, kc, uT, vT, hbiT, hbjT,
                                                        g2w, g2b, h1w, h2w, h2b, Rm);

  // 4) logits = (Q @ K^T) / sqrt(D) * R    (f32 WMMA, B transposed)
  {
    int blocks = (K_CL / 16) * (K_CL / 16) / 8;
    gemm_wmma<1, true><<<blocks, 256, 0, stream>>>(query, key_emb, attn,
                                                   K_CL, DMODEL, DMODEL, DMODEL, K_CL,
                                                   Rm, 0.0625f, nullptr);
  }
  // 5) row softmax in place
  softmax_rows<<<K_CL, 256, 0, stream>>>(attn);
  // 6) attn_out = attn @ key_emb
  {
    int blocks = (K_CL / 16) * (DMODEL / 16) / 8;
    gemm_wmma<0, false><<<blocks, 256, 0, stream>>>(attn, key_emb, aout,
                                                    DMODEL, K_CL, K_CL, DMODEL, DMODEL,
                                                    nullptr, 1.f, nullptr);
  }
  // 7) x = LN1(query + attn_out)
  add_layernorm<<<K_CL, 256, 0, stream>>>(query, aout, ln1g, ln1b, xbuf);
  // 8) ffn hidden = relu(x @ W1 + b1)   (1024 x 1024)
  {
    int blocks = (K_CL / 16) * (4 * DMODEL / 16) / 8;
    gemm_wmma<2, false><<<blocks, 256, 0, stream>>>(xbuf, f1w, ffnh,
                                                    4 * DMODEL, DMODEL, DMODEL,
                                                    4 * DMODEL, 4 * DMODEL,
                                                    nullptr, 1.f, f1b);
  }
  // 9) ffn out = hidden @ W2 + b2
  {
    int blocks = (K_CL / 16) * (DMODEL / 16) / 8;
    gemm_wmma<3, false><<<blocks, 256, 0, stream>>>(ffnh, f2w, ffno,
                                                    DMODEL, 4 * DMODEL, 4 * DMODEL,
                                                    DMODEL, DMODEL,
                                                    nullptr, 1.f, f2b);
  }
  // 10) out = LN2(x + ffn)
  add_layernorm<<<K_CL, 256, 0, stream>>>(xbuf, ffno, ln2g, ln2b, (float*)d_out);
}